// SS2D_18391049962102
// MI455X (gfx1250) — compile-verified
//
#include <hip/hip_runtime.h>
#include <hip/hip_bf16.h>

#define Bb  2
#define Hh  56
#define Ww  56
#define Cc  128
#define DIN 256
#define Nn  16
#define Rr  8
#define Kk  4
#define Ll  (Hh*Ww)      // 3136
#define E2  (2*DIN)      // 512
#define CPRJ 40          // R + 2N
#define CH  32           // scan LDS chunk

typedef float v2f __attribute__((ext_vector_type(2)));
typedef float v8f __attribute__((ext_vector_type(8)));

__device__ __forceinline__ v8f wmma_f32(v2f a, v2f b, v8f c) {
    // D = A(16x4 f32) * B(4x16 f32) + C(16x16 f32)
    return __builtin_amdgcn_wmma_f32_16x16x4_f32(false, a, false, b, (short)0, c, false, false);
}

__device__ __forceinline__ float silu_f(float v) { return v / (1.f + __expf(-v)); }

// map scan position l of direction kk -> spatial index in x0 (row-major h*W+w) layout
__device__ __forceinline__ int scan_pos(int kk, int l) {
    int ll = (kk & 2) ? (Ll - 1 - l) : l;
    if (kk & 1) { int w = ll / Hh; int h = ll - w * Hh; return h * Ww + w; }
    return ll;
}

// xz = x @ in_proj_w^T + b ; split -> xh (channel-major for conv), z = silu(z)
__global__ void k_inproj(const float* __restrict__ x, const float* __restrict__ w,
                         const float* __restrict__ bias, float* __restrict__ xh,
                         float* __restrict__ z) {
    const int wid  = blockIdx.x * (blockDim.x >> 5) + (threadIdx.x >> 5);
    const int lane = threadIdx.x & 31;
    const int nTe  = E2 / 16;                 // 32 col tiles
    const int tm   = wid / nTe;               // 0..391 row tiles
    const int te   = wid % nTe;
    if (tm >= (Bb * Ll) / 16) return;
    const int row0 = tm * 16, e0 = te * 16;
    const int hi = lane >> 4, lo = lane & 15;
    const int arow = row0 + lo;               // A: M = row
    const int ecol = e0 + lo;                 // B: N = out channel
    v8f acc = {};
    for (int kb = 0; kb < Cc; kb += 4) {
        v2f a = *(const v2f*)(x + (size_t)arow * Cc + kb + 2 * hi);
        v2f b = *(const v2f*)(w + (size_t)ecol * Cc + kb + 2 * hi);
        acc = wmma_f32(a, b, acc);
    }
    const float bv = bias[ecol];
#pragma unroll
    for (int r = 0; r < 8; ++r) {
        int row = row0 + r + 8 * hi;          // D: M = r + 8*hi
        int bb = row / Ll, l = row % Ll;
        float v = acc[r] + bv;
        if (ecol < DIN) xh[((size_t)bb * DIN + ecol) * Ll + l] = v;
        else            z[(size_t)row * DIN + (ecol - DIN)] = silu_f(v);
    }
}

// depthwise 3x3 conv (pad 1) + bias + silu; writes (b,d,l) and transposed (b,l,d)
__global__ void k_conv(const float* __restrict__ xh, const float* __restrict__ cw,
                       const float* __restrict__ cb, float* __restrict__ xc,
                       float* __restrict__ xct) {
    int idx = blockIdx.x * blockDim.x + threadIdx.x;
    if (idx >= Bb * DIN * Ll) return;
    int wq = idx % Ww; int t = idx / Ww;
    int hq = t % Hh;   t /= Hh;
    int d  = t % DIN;  int bb = t / DIN;
    const float* base = xh + ((size_t)bb * DIN + d) * Ll;
    const float* wk   = cw + d * 9;
    float acc = cb[d];
#pragma unroll
    for (int dh = -1; dh <= 1; ++dh)
#pragma unroll
        for (int dw = -1; dw <= 1; ++dw) {
            int hh = hq + dh, wp = wq + dw;
            if (hh >= 0 && hh < Hh && wp >= 0 && wp < Ww)
                acc += base[hh * Ww + wp] * wk[(dh + 1) * 3 + (dw + 1)];
        }
    float r = silu_f(acc);
    int l = hq * Ww + wq;
    xc[idx] = r;
    xct[((size_t)bb * Ll + l) * DIN + d] = r;
}

// x_dbl[b,k,c,l] = sum_d x_proj_w[k,c,d] * xs[b,k,d,l]   (xs indexed via scan_pos)
__global__ void k_xdbl(const float* __restrict__ xc, const float* __restrict__ xw,
                       float* __restrict__ dbl) {
    const int wid  = blockIdx.x * (blockDim.x >> 5) + (threadIdx.x >> 5);
    const int lane = threadIdx.x & 31;
    const int nTl = Ll / 16;                  // 196
    int t = wid;
    const int tl = t % nTl; t /= nTl;
    const int tc = t % 3;   t /= 3;           // 3 c-tiles (40 -> 48)
    const int bk = t;
    if (bk >= Bb * Kk) return;
    const int bb = bk / Kk, kk = bk % Kk;
    const int hi = lane >> 4, lo = lane & 15;
    const int c0 = tc * 16, l0 = tl * 16;
    const int ca = min(c0 + lo, CPRJ - 1);    // A row (clamped)
    const int pos = scan_pos(kk, l0 + lo);    // B col -> spatial index
    const float* xbase = xc + (size_t)bb * DIN * Ll;
    const float* wbase = xw + ((size_t)kk * CPRJ + ca) * DIN;
    v8f acc = {};
    for (int db = 0; db < DIN; db += 4) {
        int d = db + 2 * hi;
        v2f a = *(const v2f*)(wbase + d);
        v2f b;
        b.x = xbase[(size_t)(d    ) * Ll + pos];
        b.y = xbase[(size_t)(d + 1) * Ll + pos];
        acc = wmma_f32(a, b, acc);
    }
#pragma unroll
    for (int r = 0; r < 8; ++r) {
        int c = c0 + r + 8 * hi;
        if (c < CPRJ) dbl[((size_t)bk * CPRJ + c) * Ll + (l0 + lo)] = acc[r];
    }
}

// fused dt-projection + softplus + selective scan + D-skip
// writes per-direction ys[b,k,spatial,d] (deterministic; no atomics)
__global__ void __launch_bounds__(DIN)
k_scan(const float* __restrict__ xct, const float* __restrict__ dbl,
       const float* __restrict__ dtw, const float* __restrict__ dtb,
       const float* __restrict__ A_logs, const float* __restrict__ Ds,
       float* __restrict__ ys) {
    const int bk = blockIdx.x;
    const int bb = bk / Kk, kk = bk % Kk;
    const int d  = threadIdx.x;               // 0..255
    __shared__ float sBC[CH][CPRJ];
    float Av[Nn], wdt[Rr], st[Nn];
    const int kd = kk * DIN + d;
#pragma unroll
    for (int n = 0; n < Nn; ++n) { Av[n] = -__expf(A_logs[kd * Nn + n]); st[n] = 0.f; }
#pragma unroll
    for (int r = 0; r < Rr; ++r) wdt[r] = dtw[kd * Rr + r];
    const float bdt = dtb[kd], dsv = Ds[kd];
    const float* urow = xct + (size_t)bb * Ll * DIN;     // (l,d) layout
    const float* drow = dbl + (size_t)bk * CPRJ * Ll;
    float* yrow = ys + (size_t)bk * Ll * DIN;
    for (int l0 = 0; l0 < Ll; l0 += CH) {
        __syncthreads();
        for (int i = threadIdx.x; i < CPRJ * CH; i += blockDim.x) {
            int c = i / CH, j = i % CH;
            sBC[j][c] = drow[(size_t)c * Ll + l0 + j];
        }
        if (l0 + CH < Ll && threadIdx.x < CPRJ)
            __builtin_prefetch(drow + (size_t)threadIdx.x * Ll + l0 + CH, 0, 1);
        __syncthreads();
#pragma unroll 1
        for (int j = 0; j < CH; ++j) {
            int l = l0 + j;
            float t = bdt;
#pragma unroll
            for (int r = 0; r < Rr; ++r) t += wdt[r] * sBC[j][r];
            float delta = (t > 20.f) ? t : logf(1.f + __expf(t));
            int pos = scan_pos(kk, l);
            float u  = urow[(size_t)pos * DIN + d];       // coalesced across lanes
            float du = delta * u;
            float yv = dsv * u;
#pragma unroll
            for (int n = 0; n < Nn; ++n) {
                float s = __expf(delta * Av[n]) * st[n] + du * sBC[j][Rr + n];
                st[n] = s;
                yv += s * sBC[j][Rr + Nn + n];
            }
            yrow[(size_t)pos * DIN + d] = yv;             // coalesced across lanes
        }
    }
}

// sum 4 directions (fixed order) + LayerNorm over d + * silu(z) -> yn[b,l,d]
__global__ void __launch_bounds__(DIN)
k_norm(const float* __restrict__ ys, const float* __restrict__ onw,
       const float* __restrict__ onb, const float* __restrict__ z,
       float* __restrict__ yn) {
    const int row = blockIdx.x, d = threadIdx.x;          // row = b*L + l
    const int bb = row / Ll, l = row % Ll;
    __shared__ float r1[DIN], r2[DIN];
    float v = 0.f;
#pragma unroll
    for (int k = 0; k < Kk; ++k)
        v += ys[(((size_t)(bb * Kk + k)) * Ll + l) * DIN + d];
    r1[d] = v; r2[d] = v * v;
    __syncthreads();
    for (int s = DIN / 2; s > 0; s >>= 1) {
        if (d < s) { r1[d] += r1[d + s]; r2[d] += r2[d + s]; }
        __syncthreads();
    }
    float mu  = r1[0] * (1.f / DIN);
    float var = r2[0] * (1.f / DIN) - mu * mu;
    float o = (v - mu) * rsqrtf(var + 1e-5f) * onw[d] + onb[d];
    yn[(size_t)row * DIN + d] = o * z[(size_t)row * DIN + d];
}

// out = yn @ out_proj_w^T + b
__global__ void k_outproj(const float* __restrict__ yn, const float* __restrict__ w,
                          const float* __restrict__ bias, float* __restrict__ out) {
    const int wid  = blockIdx.x * (blockDim.x >> 5) + (threadIdx.x >> 5);
    const int lane = threadIdx.x & 31;
    const int nTc = Cc / 16;                  // 8
    const int tm = wid / nTc, tc = wid % nTc;
    if (tm >= (Bb * Ll) / 16) return;
    const int row0 = tm * 16, c0 = tc * 16;
    const int hi = lane >> 4, lo = lane & 15;
    v8f acc = {};
    for (int db = 0; db < DIN; db += 4) {
        int dd = db + 2 * hi;
        v2f a = *(const v2f*)(yn + (size_t)(row0 + lo) * DIN + dd);
        v2f b = *(const v2f*)(w  + (size_t)(c0 + lo) * DIN + dd);
        acc = wmma_f32(a, b, acc);
    }
    const float bv = bias[c0 + lo];
#pragma unroll
    for (int r = 0; r < 8; ++r) {
        int row = row0 + r + 8 * hi;
        out[(size_t)row * Cc + (c0 + lo)] = acc[r] + bv;
    }
}

extern "C" void kernel_launch(void* const* d_in, const int* in_sizes, int n_in,
                              void* d_out, int out_size, void* d_ws, size_t ws_size,
                              hipStream_t stream) {
    const float* x    = (const float*)d_in[0];
    const float* ipw  = (const float*)d_in[1];
    const float* ipb  = (const float*)d_in[2];
    const float* cw   = (const float*)d_in[3];
    const float* cb   = (const float*)d_in[4];
    const float* xpw  = (const float*)d_in[5];
    const float* dtw  = (const float*)d_in[6];
    const float* dtb  = (const float*)d_in[7];
    const float* alog = (const float*)d_in[8];
    const float* ds   = (const float*)d_in[9];
    const float* onw  = (const float*)d_in[10];
    const float* onb  = (const float*)d_in[11];
    const float* opw  = (const float*)d_in[12];
    const float* opb  = (const float*)d_in[13];
    float* out = (float*)d_out;

    float* ws  = (float*)d_ws;
    float* xh  = ws;                                   // B*DIN*L (reused as yn later)
    float* zb  = xh  + (size_t)Bb * DIN * Ll;          // B*L*DIN
    float* xcb = zb  + (size_t)Bb * Ll * DIN;          // B*DIN*L  (d-major, for x_dbl)
    float* xct = xcb + (size_t)Bb * DIN * Ll;          // B*L*DIN  (l-major, for scan)
    float* dbl = xct + (size_t)Bb * Ll * DIN;          // B*K*40*L
    float* ysb = dbl + (size_t)Bb * Kk * CPRJ * Ll;    // B*K*L*DIN (per-direction)
    float* yn  = xh;                                   // alias: xh dead after conv

    const int nElem = Bb * DIN * Ll;
    // in_proj: 392 row-tiles * 32 col-tiles = 12544 waves -> 1568 blocks of 8 waves
    k_inproj<<<1568, 256, 0, stream>>>(x, ipw, ipb, xh, zb);
    k_conv<<<(nElem + 255) / 256, 256, 0, stream>>>(xh, cw, cb, xcb, xct);
    // x_dbl: 8 (b,k) * 3 c-tiles * 196 l-tiles = 4704 waves -> 588 blocks
    k_xdbl<<<588, 256, 0, stream>>>(xcb, xpw, dbl);
    k_scan<<<Bb * Kk, DIN, 0, stream>>>(xct, dbl, dtw, dtb, alog, ds, ysb);
    k_norm<<<Bb * Ll, DIN, 0, stream>>>(ysb, onw, onb, zb, yn);
    // out_proj: 392 row-tiles * 8 col-tiles = 3136 waves -> 392 blocks
    k_outproj<<<392, 256, 0, stream>>>(yn, opw, opb, out);
}